// lstm_841813590590
// MI455X (gfx1250) — compile-verified
//
#include <hip/hip_runtime.h>
#include <hip/hip_bf16.h>

typedef __attribute__((ext_vector_type(2))) float v2f;
typedef __attribute__((ext_vector_type(8))) float v8f;

#define B_TOT 4096
#define S_LEN 512
#define D_IN  8
#define NT    2   // independent 16-batch WMMA tiles per wave (ILP)

static __device__ __forceinline__ v8f wmma4(v2f a, v2f b, v8f c) {
    // V_WMMA_F32_16X16X4_F32 : D(16x16,f32) = A(16x4,f32) * B(4x16,f32) + C
    return __builtin_amdgcn_wmma_f32_16x16x4_f32(
        /*neg_a=*/false, a, /*neg_b=*/false, b,
        /*c_mod=*/(short)0, c, /*reuse_a=*/false, /*reuse_b=*/false);
}

// Gate-row permutation: A/D row m holds original gate perm(m).
// Rows: [i0,i1,f0,f1,g0,g1,o0,o1 | i2,i3,f2,f3,g2,g3,o2,o3]
// -> D tile: each lane owns all 8 gates of its 2 cells; no cross-lane exchange.
static __device__ __forceinline__ int gperm(int m) {
    return ((m >> 1) & 3) * 4 + (m >> 3) * 2 + (m & 1);
}

static __device__ __forceinline__ float fsig(float x) {
    float e = __builtin_amdgcn_exp2f(-x * 1.4426950408889634f);
    return __builtin_amdgcn_rcpf(1.0f + e);   // saturates 0/1 at -/+inf
}
static __device__ __forceinline__ float ftanh(float x) {
    float e = __builtin_amdgcn_exp2f(x * 2.8853900817779268f);
    return 1.0f - 2.0f * __builtin_amdgcn_rcpf(e + 1.0f);  // saturates -1/+1, no NaN
}

// d (permuted rows): d0,d1=i | d2,d3=f | d4,d5=g | d6,d7=o for this lane's 2 cells.
// Returns new h pair directly in B-matrix layout.
static __device__ __forceinline__ v2f lstm_update(v8f d, float& cA, float& cB) {
    cA = fsig(d[2]) * cA + fsig(d[0]) * ftanh(d[4]);
    cB = fsig(d[3]) * cB + fsig(d[1]) * ftanh(d[5]);
    v2f h;
    h.x = fsig(d[6]) * ftanh(cA);
    h.y = fsig(d[7]) * ftanh(cB);
    return h;
}

static __device__ __forceinline__ float swap16(float v) {
    // ds_swizzle SWAPX16: xor_mask=0x10, and_mask=0x1f -> imm 0x401F
    return __int_as_float(__builtin_amdgcn_ds_swizzle(__float_as_int(v), 0x401F));
}

__global__ __launch_bounds__(32) void lstm4_wmma_kernel(
    const float* __restrict__ x,
    const float* __restrict__ w_ih0, const float* __restrict__ w_hh0,
    const float* __restrict__ b_ih0, const float* __restrict__ b_hh0,
    const float* __restrict__ w_ih,  const float* __restrict__ w_hh,
    const float* __restrict__ b_ih,  const float* __restrict__ b_hh,
    const float* __restrict__ w_out, const float* __restrict__ b_out,
    float* __restrict__ out)
{
    const int  lane  = threadIdx.x;
    const int  m     = lane & 15;          // A row slot / B,C,D column (batch)
    const bool lower = lane < 16;
    const int  kh    = lower ? 0 : 2;      // K sub-rows held by this half-wave
    const int  gofs  = lower ? 0 : 8;      // D-row block for this half-wave
    const int  pm    = gperm(m);           // permuted gate row loaded by this lane
    const int  b0    = blockIdx.x * (16 * NT);

    // ---- resident weights in WMMA A-layout with permuted gate rows ----
    v2f a_ih0_lo = { w_ih0[pm * 8 + kh],     w_ih0[pm * 8 + kh + 1] };      // K=0..3
    v2f a_ih0_hi = { w_ih0[pm * 8 + 4 + kh], w_ih0[pm * 8 + 4 + kh + 1] };  // K=4..7
    v2f a_hh0    = { w_hh0[pm * 4 + kh],     w_hh0[pm * 4 + kh + 1] };
    v2f a_ihL[3], a_hhL[3];
    v8f cb0, cbL[3];
#pragma unroll
    for (int r = 0; r < 8; ++r) {
        const int g = gperm(r + gofs);
        cb0[r] = b_ih0[g] + b_hh0[g];
    }
#pragma unroll
    for (int L = 0; L < 3; ++L) {
        a_ihL[L].x = w_ih[L * 64 + pm * 4 + kh];
        a_ihL[L].y = w_ih[L * 64 + pm * 4 + kh + 1];
        a_hhL[L].x = w_hh[L * 64 + pm * 4 + kh];
        a_hhL[L].y = w_hh[L * 64 + pm * 4 + kh + 1];
#pragma unroll
        for (int r = 0; r < 8; ++r) {
            const int g = gperm(r + gofs);
            cbL[L][r] = b_ih[L * 16 + g] + b_hh[L * 16 + g];
        }
    }
    const float woA = w_out[lower ? 0 : 2];
    const float woB = w_out[lower ? 1 : 3];
    const float bo  = b_out[0];

    // ---- recurrent state (registers): h in B-layout, c split across half-waves ----
    v2f   bh[NT][4];
    float cA[NT][4], cB[NT][4];
#pragma unroll
    for (int q = 0; q < NT; ++q)
#pragma unroll
        for (int L = 0; L < 4; ++L) {
            bh[q][L].x = 0.f; bh[q][L].y = 0.f;
            cA[q][L] = 0.f;   cB[q][L] = 0.f;
        }

    const float* xp[NT];
    size_t       op[NT];
#pragma unroll
    for (int q = 0; q < NT; ++q) {
        const size_t b = (size_t)(b0 + q * 16 + m);
        xp[q] = x + b * (size_t)S_LEN * D_IN + kh;   // this lane's K sub-rows of X^T
        op[q] = b * (size_t)S_LEN;
    }

    // software pipeline: preload t=0 x tiles
    v2f nx0[NT], nx1[NT];
#pragma unroll
    for (int q = 0; q < NT; ++q) {
        nx0[q].x = xp[q][0]; nx0[q].y = xp[q][1];
        nx1[q].x = xp[q][4]; nx1[q].y = xp[q][5];
    }

    for (int t = 0; t < S_LEN; ++t) {
        v2f bx0[NT], bx1[NT];
#pragma unroll
        for (int q = 0; q < NT; ++q) { bx0[q] = nx0[q]; bx1[q] = nx1[q]; }

        // prefetch next timestep (clamped; last iter harmlessly re-reads t=S-1)
        const int tn = (t + 1 < S_LEN) ? (t + 1) : (S_LEN - 1);
#pragma unroll
        for (int q = 0; q < NT; ++q) {
            const float* p = xp[q] + (size_t)tn * D_IN;
            nx0[q].x = p[0]; nx0[q].y = p[1];
            nx1[q].x = p[4]; nx1[q].y = p[5];
        }

#pragma unroll
        for (int q = 0; q < NT; ++q) {
            // layer 0: gates = Wih0 @ x_t + Whh0 @ h + (b_ih0 + b_hh0)
            v8f acc = wmma4(a_ih0_lo, bx0[q], cb0);
            acc = wmma4(a_ih0_hi, bx1[q], acc);
            acc = wmma4(a_hh0, bh[q][0], acc);
            bh[q][0] = lstm_update(acc, cA[q][0], cB[q][0]);

            // layers 1..3: input is previous layer's fresh h (already B-layout)
#pragma unroll
            for (int L = 0; L < 3; ++L) {
                v8f a2 = wmma4(a_ihL[L], bh[q][L], cbL[L]);
                a2 = wmma4(a_hhL[L], bh[q][L + 1], a2);
                bh[q][L + 1] = lstm_update(a2, cA[q][L + 1], cB[q][L + 1]);
            }

            // linear head: y = w_out . h3 + b_out (each half holds 2 of 4 terms)
            float pr = woA * bh[q][3].x + woB * bh[q][3].y;
            pr += swap16(pr);
            if (lower) out[op[q] + t] = pr + bo;
        }
    }
}

extern "C" void kernel_launch(void* const* d_in, const int* in_sizes, int n_in,
                              void* d_out, int out_size, void* d_ws, size_t ws_size,
                              hipStream_t stream) {
    (void)in_sizes; (void)n_in; (void)d_ws; (void)ws_size; (void)out_size;
    const float* x     = (const float*)d_in[0];
    const float* w_ih0 = (const float*)d_in[1];
    const float* w_hh0 = (const float*)d_in[2];
    const float* b_ih0 = (const float*)d_in[3];
    const float* b_hh0 = (const float*)d_in[4];
    const float* w_ih  = (const float*)d_in[5];
    const float* w_hh  = (const float*)d_in[6];
    const float* b_ih  = (const float*)d_in[7];
    const float* b_hh  = (const float*)d_in[8];
    const float* w_out = (const float*)d_in[9];
    const float* b_out = (const float*)d_in[10];
    float* out = (float*)d_out;

    dim3 grid(B_TOT / (16 * NT));  // 128 waves, 32 batch columns each
    dim3 block(32);
    lstm4_wmma_kernel<<<grid, block, 0, stream>>>(
        x, w_ih0, w_hh0, b_ih0, b_hh0, w_ih, w_hh, b_ih, b_hh, w_out, b_out, out);
}